// ProxyDeltaLinearV68_78245714199351
// MI455X (gfx1250) — compile-verified
//
#include <hip/hip_runtime.h>
#include <hip/hip_bf16.h>
#include <cstdint>
#include <cstddef>

// ---------------------------------------------------------------------------
// ProxyDeltaLinear forward for MI455X (gfx1250).
// Forward value reduces to the L1 path only (stop_gradient cancels the L2
// GEMM term), so the core is a VALU-bound L1 cdist with:
//   * tanh precomputed once (kernel 1)
//   * async global->LDS tile staging (CDNA5 GLOBAL_LOAD_ASYNC_TO_LDS_B128)
//   * 64x64 tiles, 4x4 register micro-tiles, b128 LDS reads (kernel 2)
//   * gain column-scaling as T x diag(g) on the WMMA pipe (kernel 3)
// ---------------------------------------------------------------------------

#define AS1 __attribute__((address_space(1)))
#define AS3 __attribute__((address_space(3)))

typedef __attribute__((ext_vector_type(2))) float v2f;
typedef __attribute__((ext_vector_type(8))) float v8f;
typedef __attribute__((ext_vector_type(4))) int   v4i;

#define DIN   256   // Din (contraction length)
#define DOUT  256   // Dout (output columns / weight rows)
#define KC    32    // k-chunk staged in LDS per iteration
#define KPAD  36    // padded LDS row pitch in floats (keeps 16B alignment)
#define BT    64    // block tile edge (64 rows x 64 cols per workgroup)

// ---- CDNA5 async copy helpers (graceful fallback keeps compile green) -----
__device__ __forceinline__ void async_copy16(void* lds, const void* g) {
#if __has_builtin(__builtin_amdgcn_global_load_async_to_lds_b128)
  // Builtin signature (probe-confirmed by the round-1 diagnostic):
  //   (v4i addrspace(1)*, v4i addrspace(3)*, imm offset, imm cpol)
  __builtin_amdgcn_global_load_async_to_lds_b128(
      (AS1 v4i*)(void*)g, (AS3 v4i*)lds, 0, 0);
#else
  *(float4*)lds = *(const float4*)g;
#endif
}

__device__ __forceinline__ void wait_async_all() {
#if __has_builtin(__builtin_amdgcn_global_load_async_to_lds_b128)
#if __has_builtin(__builtin_amdgcn_s_wait_asynccnt)
  __builtin_amdgcn_s_wait_asynccnt(0);
#else
  asm volatile("s_wait_asynccnt 0" ::: "memory");
#endif
#endif
}

// ---------------------------------------------------------------------------
// Kernel 1: elementwise tanh of x and weight into workspace (vectorized x4).
// ---------------------------------------------------------------------------
__global__ __launch_bounds__(256) void tanh_prep(
    const float* __restrict__ x, const float* __restrict__ w,
    float* __restrict__ xact, float* __restrict__ wact, int nx4, int nw4) {
  int i = blockIdx.x * blockDim.x + threadIdx.x;
  if (i < nx4) {
    float4 v = ((const float4*)x)[i];
    float4 r;
    r.x = tanhf(v.x); r.y = tanhf(v.y); r.z = tanhf(v.z); r.w = tanhf(v.w);
    ((float4*)xact)[i] = r;
  } else {
    int j = i - nx4;
    if (j < nw4) {
      float4 v = ((const float4*)w)[j];
      float4 r;
      r.x = tanhf(v.x); r.y = tanhf(v.y); r.z = tanhf(v.z); r.w = tanhf(v.w);
      ((float4*)wact)[j] = r;
    }
  }
}

// ---------------------------------------------------------------------------
// Kernel 2: L1 cdist core. Each block computes a 64x64 (n,d) tile; each of
// the 256 threads owns a 4x4 micro-tile. k is chunked by 32 through LDS.
// Writes (0.5 - l1_mean) to out; gain scaling happens in kernel 3 (WMMA).
// ---------------------------------------------------------------------------
__global__ __launch_bounds__(256) void l1_tile_kernel(
    const float* __restrict__ xact,   // (N, 256) tanh(x)
    const float* __restrict__ wact,   // (256, 256) tanh(w)
    float* __restrict__ out) {        // (N, 256)
  __shared__ __align__(16) float xs[BT * KPAD];
  __shared__ __align__(16) float wsh[BT * KPAD];

  const int t  = threadIdx.x;
  const int tx = t & 15;    // column group (d)
  const int ty = t >> 4;    // row group (n)
  const int n0 = blockIdx.x * BT;
  const int d0 = blockIdx.y * BT;

  float acc[4][4] = {};

  for (int k0 = 0; k0 < DIN; k0 += KC) {
    // Stage 64x32 x-tile and 64x32 w-tile: 512 16B segments each,
    // 2 per thread per tile -> 4 async-to-LDS instructions per thread.
#pragma unroll
    for (int s = t; s < 512; s += 256) {
      const int row = s >> 3;        // 0..63
      const int seg = s & 7;         // 0..7 (16B segments of the 32-float row)
      async_copy16(&xs[row * KPAD + seg * 4],
                   xact + (size_t)(n0 + row) * DIN + k0 + seg * 4);
      async_copy16(&wsh[row * KPAD + seg * 4],
                   wact + (size_t)(d0 + row) * DIN + k0 + seg * 4);
    }
    wait_async_all();
    __syncthreads();

#pragma unroll
    for (int k4 = 0; k4 < KC / 4; ++k4) {
      float4 xv[4], wv[4];
#pragma unroll
      for (int i = 0; i < 4; ++i)
        xv[i] = *(const float4*)&xs[(ty * 4 + i) * KPAD + k4 * 4];
#pragma unroll
      for (int j = 0; j < 4; ++j)
        wv[j] = *(const float4*)&wsh[(tx * 4 + j) * KPAD + k4 * 4];
#pragma unroll
      for (int i = 0; i < 4; ++i)
#pragma unroll
        for (int j = 0; j < 4; ++j) {
          // v_sub_f32 + v_add_f32 with |src| modifier: 2 VALU ops/element
          acc[i][j] += fabsf(xv[i].x - wv[j].x);
          acc[i][j] += fabsf(xv[i].y - wv[j].y);
          acc[i][j] += fabsf(xv[i].z - wv[j].z);
          acc[i][j] += fabsf(xv[i].w - wv[j].w);
        }
    }
    __syncthreads();
  }

  const float inv = 1.0f / (float)DIN;
#pragma unroll
  for (int i = 0; i < 4; ++i) {
    float4 o;
    o.x = 0.5f - acc[i][0] * inv;
    o.y = 0.5f - acc[i][1] * inv;
    o.z = 0.5f - acc[i][2] * inv;
    o.w = 0.5f - acc[i][3] * inv;
    *(float4*)(out + (size_t)(n0 + ty * 4 + i) * DOUT + d0 + tx * 4) = o;
  }
}

// ---------------------------------------------------------------------------
// Kernel 3: out = out x diag(gain), exactly, on the matrix pipe.
// Per 16x16 tile: 4 chained V_WMMA_F32_16X16X4_F32 with B = 4x16 slices of
// the diagonal gain block. In-place on d_out (tile-local read-then-write).
// Fragment layouts per ISA 7.12.2:
//   A 16x4 : lane l -> M = l%16 ; VGPR0 = K(2*(l/16)), VGPR1 = K(2*(l/16)+1)
//   B 4x16 : lane l -> N = l%16 ; VGPR0 = K(2*(l/16)), VGPR1 = K(2*(l/16)+1)
//   C/D    : lane l, VGPR r -> M = r + 8*(l/16), N = l%16
// ---------------------------------------------------------------------------
__global__ __launch_bounds__(256) void gain_scale_wmma(
    float* __restrict__ out, const float* __restrict__ gain, int ntd) {
  const int lane = threadIdx.x & 31;
  const int wave = threadIdx.x >> 5;
  const int tile = blockIdx.x * 8 + wave;   // 8 waves per block, 1 tile each
  const int tn = tile / ntd;
  const int td = tile % ntd;
  const int n0 = tn * 16;
  const int d0 = td * 16;

  const int m    = lane & 15;   // A row index; also B/C/D column index
  const int half = lane >> 4;

  const float g = gain[d0 + m];

  v8f c = {};
#pragma unroll
  for (int ch = 0; ch < 4; ++ch) {
    const int k0 = ch * 4;
    const float* row = out + (size_t)(n0 + m) * DOUT + d0 + k0;
    v2f a;
    a.x = row[2 * half];
    a.y = row[2 * half + 1];
    v2f b;  // diagonal gain block rows k0+2*half, k0+2*half+1
    b.x = (k0 + 2 * half     == m) ? g : 0.0f;
    b.y = (k0 + 2 * half + 1 == m) ? g : 0.0f;
    c = __builtin_amdgcn_wmma_f32_16x16x4_f32(
        /*neg_a=*/false, a, /*neg_b=*/false, b,
        /*c_mod=*/(short)0, c, /*reuse_a=*/false, /*reuse_b=*/false);
  }
#pragma unroll
  for (int r = 0; r < 8; ++r)
    out[(size_t)(n0 + r + 8 * half) * DOUT + d0 + m] = c[r];
}

// ---------------------------------------------------------------------------
extern "C" void kernel_launch(void* const* d_in, const int* in_sizes, int n_in,
                              void* d_out, int out_size, void* d_ws, size_t ws_size,
                              hipStream_t stream) {
  const float* x    = (const float*)d_in[0];  // (B,T,256) f32
  const float* w    = (const float*)d_in[1];  // (256,256) f32
  const float* gain = (const float*)d_in[2];  // (1,256,1) f32
  float* out = (float*)d_out;                 // (B,T,256) f32

  const int nx = in_sizes[0];                 // B*T*256
  const int nw = in_sizes[1];                 // 256*256
  const int N  = nx / DIN;                    // B*T = 2048

  float* xact = (float*)d_ws;                 // N*256 floats (2 MB)
  float* wact = xact + nx;                    // 256*256 floats (256 KB)

  // 1) tanh activations into workspace
  const int nx4 = nx / 4, nw4 = nw / 4;
  const int total4 = nx4 + nw4;
  tanh_prep<<<(total4 + 255) / 256, 256, 0, stream>>>(x, w, xact, wact, nx4, nw4);

  // 2) L1 cdist core -> out = 0.5 - l1_mean
  dim3 g2(N / BT, DOUT / BT);
  l1_tile_kernel<<<g2, 256, 0, stream>>>(xact, wact, out);

  // 3) column scale by gain via WMMA (in-place), 8 tiles (waves) per block
  const int ntd = DOUT / 16;                  // 16
  const int tiles = (N / 16) * ntd;           // 2048
  gain_scale_wmma<<<tiles / 8, 256, 0, stream>>>(out, gain, ntd);
}